// DecoderPolicyGradient_47253230191315
// MI455X (gfx1250) — compile-verified
//
#include <hip/hip_runtime.h>
#include <hip/hip_bf16.h>
#include <math.h>

// Problem constants (match reference)
#define V_   10000
#define E_   512
#define H_   1024
#define B_   256
#define T_   20
#define G4H  4096   // 4*H

typedef float v2f __attribute__((ext_vector_type(2)));
typedef float v8f __attribute__((ext_vector_type(8)));

#if defined(__HIP_DEVICE_COMPILE__)
#if !__has_builtin(__builtin_amdgcn_wmma_f32_16x16x4_f32)
#error "gfx1250 fp32 WMMA builtin not available"
#endif
#endif

__device__ __forceinline__ v8f wmma4(v2f a, v2f b, v8f c) {
  // (neg_a, A, neg_b, B, c_mod, C, reuse_a, reuse_b)
  return __builtin_amdgcn_wmma_f32_16x16x4_f32(false, a, false, b, (short)0, c,
                                               false, false);
}

// ---------------------------------------------------------------------------
// Packing offsets: lengths sorted descending -> b_idx for time t is 0..cnt_t-1
// ---------------------------------------------------------------------------
__global__ void pack_offsets_kernel(const int* __restrict__ lengths,
                                    int* __restrict__ cnts,
                                    int* __restrict__ offs) {
  if (threadIdx.x == 0 && blockIdx.x == 0) {
    int off = 0;
    for (int t = 0; t < T_; ++t) {
      int c = 0;
      for (int b = 0; b < B_; ++b) c += (lengths[b] > t) ? 1 : 0;
      cnts[t] = c;
      offs[t] = off;
      off += c;
    }
  }
}

// ---------------------------------------------------------------------------
// GEMM1: Gx[t*B+b][0:4H] = xs[b][t][:] @ W_ih.T + (b_ih + b_hh)
//   xs row: t==0 -> features[b], else embed_W[captions[b][t-1]]
// Block: 256 thr = 8 waves (2M x 4N). Wave tile 32x64. K = E = 512.
// ---------------------------------------------------------------------------
__global__ __launch_bounds__(256) void gemm1_xproj_kernel(
    const float* __restrict__ features, const int* __restrict__ captions,
    const float* __restrict__ embed_W, const float* __restrict__ W_ih,
    const float* __restrict__ b_ih, const float* __restrict__ b_hh,
    float* __restrict__ Gx) {
  const int lane = threadIdx.x & 31;
  const int wave = threadIdx.x >> 5;
  const int wm = wave >> 2, wn = wave & 3;
  const int half = lane >> 4, lm = lane & 15;
  const int m0 = blockIdx.y * 64 + wm * 32;
  const int n0 = blockIdx.x * 256 + wn * 64;
  const int koff = 2 * half;

  // Gathered A row pointers (embedding fused into GEMM)
  const float* aptr[2];
#pragma unroll
  for (int i = 0; i < 2; ++i) {
    int row = m0 + i * 16 + lm;    // row = t*256 + b (64-row tile stays in one t)
    int t = row >> 8;
    int b = row & 255;
    aptr[i] = (t == 0) ? (features + (size_t)b * E_)
                       : (embed_W + (size_t)captions[b * T_ + (t - 1)] * E_);
  }
  const float* bptr[4];
#pragma unroll
  for (int j = 0; j < 4; ++j)
    bptr[j] = W_ih + (size_t)(n0 + j * 16 + lm) * E_;

  v8f acc[2][4] = {};

#pragma unroll 2
  for (int k0 = 0; k0 < E_; k0 += 4) {
    v2f a[2], bf[4];
#pragma unroll
    for (int i = 0; i < 2; ++i) a[i] = *(const v2f*)(aptr[i] + k0 + koff);
#pragma unroll
    for (int j = 0; j < 4; ++j) bf[j] = *(const v2f*)(bptr[j] + k0 + koff);
#pragma unroll
    for (int i = 0; i < 2; ++i)
#pragma unroll
      for (int j = 0; j < 4; ++j) acc[i][j] = wmma4(a[i], bf[j], acc[i][j]);
  }

#pragma unroll
  for (int j = 0; j < 4; ++j) {
    int col = n0 + j * 16 + lm;
    float bias = b_ih[col] + b_hh[col];
#pragma unroll
    for (int i = 0; i < 2; ++i)
#pragma unroll
      for (int r = 0; r < 8; ++r) {
        int row = m0 + i * 16 + half * 8 + r;
        Gx[(size_t)row * G4H + col] = acc[i][j][r] + bias;
      }
  }
}

// ---------------------------------------------------------------------------
// GEMM2 (per timestep): Gt[b][0:4H] += h_cur[b][:] @ W_hh.T   (in place)
// M=256, N=4096, K=1024
// ---------------------------------------------------------------------------
__global__ __launch_bounds__(256) void gemm2_hproj_kernel(
    const float* __restrict__ h_cur, const float* __restrict__ W_hh,
    float* __restrict__ Gt) {
  const int lane = threadIdx.x & 31;
  const int wave = threadIdx.x >> 5;
  const int wm = wave >> 2, wn = wave & 3;
  const int half = lane >> 4, lm = lane & 15;
  const int m0 = blockIdx.y * 64 + wm * 32;
  const int n0 = blockIdx.x * 256 + wn * 64;
  const int koff = 2 * half;

  const float* aptr[2];
#pragma unroll
  for (int i = 0; i < 2; ++i) aptr[i] = h_cur + (size_t)(m0 + i * 16 + lm) * H_;
  const float* bptr[4];
#pragma unroll
  for (int j = 0; j < 4; ++j)
    bptr[j] = W_hh + (size_t)(n0 + j * 16 + lm) * H_;

  v8f acc[2][4] = {};

#pragma unroll 2
  for (int k0 = 0; k0 < H_; k0 += 4) {
    v2f a[2], bf[4];
#pragma unroll
    for (int i = 0; i < 2; ++i) a[i] = *(const v2f*)(aptr[i] + k0 + koff);
#pragma unroll
    for (int j = 0; j < 4; ++j) bf[j] = *(const v2f*)(bptr[j] + k0 + koff);
#pragma unroll
    for (int i = 0; i < 2; ++i)
#pragma unroll
      for (int j = 0; j < 4; ++j) acc[i][j] = wmma4(a[i], bf[j], acc[i][j]);
  }

#pragma unroll
  for (int j = 0; j < 4; ++j) {
    int col = n0 + j * 16 + lm;
#pragma unroll
    for (int i = 0; i < 2; ++i)
#pragma unroll
      for (int r = 0; r < 8; ++r) {
        int row = m0 + i * 16 + half * 8 + r;
        size_t idx = (size_t)row * G4H + col;
        Gt[idx] = Gt[idx] + acc[i][j][r];   // accumulate onto x-projection
      }
  }
}

// ---------------------------------------------------------------------------
// LSTM pointwise gates: i,f,g,o split along 4H; update c,h; emit hs[t]
// ---------------------------------------------------------------------------
__global__ __launch_bounds__(256) void lstm_pointwise_kernel(
    const float* __restrict__ Gt, float* __restrict__ h_cur,
    float* __restrict__ c_cur, float* __restrict__ hs_t) {
  int idx = blockIdx.x * blockDim.x + threadIdx.x;
  if (idx >= B_ * H_) return;
  int b = idx >> 10;          // / H
  int h = idx & (H_ - 1);
  const float* g = Gt + (size_t)b * G4H;
  float gi = g[h];
  float gf = g[H_ + h];
  float gg = g[2 * H_ + h];
  float go = g[3 * H_ + h];
  float si = 1.f / (1.f + __expf(-gi));
  float sf = 1.f / (1.f + __expf(-gf));
  float so = 1.f / (1.f + __expf(-go));
  float c = sf * c_cur[idx] + si * tanhf(gg);
  float hh = so * tanhf(c);
  c_cur[idx] = c;
  h_cur[idx] = hh;
  hs_t[idx] = hh;
}

// ---------------------------------------------------------------------------
// GEMM3: out[offs[t]+b][v] = hs[t*B+b][:] @ lin_W.T + lin_b   (b < cnts[t])
// M=5120, N=10000 (clamped), K=1024
// ---------------------------------------------------------------------------
__global__ __launch_bounds__(256) void gemm3_vocab_kernel(
    const float* __restrict__ hs, const float* __restrict__ lin_W,
    const float* __restrict__ lin_b, const int* __restrict__ cnts,
    const int* __restrict__ offs, float* __restrict__ out) {
  const int lane = threadIdx.x & 31;
  const int wave = threadIdx.x >> 5;
  const int wm = wave >> 2, wn = wave & 3;
  const int half = lane >> 4, lm = lane & 15;
  const int m0 = blockIdx.y * 64 + wm * 32;
  const int n0 = blockIdx.x * 256 + wn * 64;
  const int koff = 2 * half;

  const float* aptr[2];
#pragma unroll
  for (int i = 0; i < 2; ++i) aptr[i] = hs + (size_t)(m0 + i * 16 + lm) * H_;
  const float* bptr[4];
#pragma unroll
  for (int j = 0; j < 4; ++j) {
    int col = n0 + j * 16 + lm;
    if (col > V_ - 1) col = V_ - 1;   // clamp loads; store is predicated
    bptr[j] = lin_W + (size_t)col * H_;
  }

  v8f acc[2][4] = {};

#pragma unroll 2
  for (int k0 = 0; k0 < H_; k0 += 4) {
    v2f a[2], bf[4];
#pragma unroll
    for (int i = 0; i < 2; ++i) a[i] = *(const v2f*)(aptr[i] + k0 + koff);
#pragma unroll
    for (int j = 0; j < 4; ++j) bf[j] = *(const v2f*)(bptr[j] + k0 + koff);
#pragma unroll
    for (int i = 0; i < 2; ++i)
#pragma unroll
      for (int j = 0; j < 4; ++j) acc[i][j] = wmma4(a[i], bf[j], acc[i][j]);
  }

#pragma unroll
  for (int i = 0; i < 2; ++i) {
    int t = (m0 + i * 16) >> 8;      // constant within 16-row fragment
    int cnt = cnts[t];
    int off = offs[t];
#pragma unroll
    for (int j = 0; j < 4; ++j) {
      int col = n0 + j * 16 + lm;
      if (col < V_) {
        float bias = lin_b[col];
#pragma unroll
        for (int r = 0; r < 8; ++r) {
          int row = m0 + i * 16 + half * 8 + r;
          int b = row & 255;
          if (b < cnt)
            out[(size_t)(off + b) * V_ + col] = acc[i][j][r] + bias;
        }
      }
    }
  }
}

// ---------------------------------------------------------------------------
extern "C" void kernel_launch(void* const* d_in, const int* in_sizes, int n_in,
                              void* d_out, int out_size, void* d_ws,
                              size_t ws_size, hipStream_t stream) {
  const float* features = (const float*)d_in[0];
  const int*   captions = (const int*)d_in[1];
  const float* h0       = (const float*)d_in[2];
  const float* c0       = (const float*)d_in[3];
  /* d_in[4] = maxlen scalar (compile-time T_) */
  const int*   lengths  = (const int*)d_in[5];
  const float* embed_W  = (const float*)d_in[6];
  const float* W_ih     = (const float*)d_in[7];
  const float* W_hh     = (const float*)d_in[8];
  const float* b_ih     = (const float*)d_in[9];
  const float* b_hh     = (const float*)d_in[10];
  const float* lin_W    = (const float*)d_in[11];
  const float* lin_b    = (const float*)d_in[12];
  float* out = (float*)d_out;

  // Workspace layout
  float* Gx    = (float*)d_ws;                       // T*B*4H  (83.9 MB)
  float* hs    = Gx + (size_t)T_ * B_ * G4H;         // T*B*H   (21.0 MB)
  float* h_cur = hs + (size_t)T_ * B_ * H_;          // B*H
  float* c_cur = h_cur + (size_t)B_ * H_;            // B*H
  int*   cnts  = (int*)(c_cur + (size_t)B_ * H_);    // T
  int*   offs  = cnts + T_;                          // T

  pack_offsets_kernel<<<1, 32, 0, stream>>>(lengths, cnts, offs);
  (void)hipMemcpyAsync(h_cur, h0, (size_t)B_ * H_ * sizeof(float),
                       hipMemcpyDeviceToDevice, stream);
  (void)hipMemcpyAsync(c_cur, c0, (size_t)B_ * H_ * sizeof(float),
                       hipMemcpyDeviceToDevice, stream);

  dim3 blk(256);
  // Input projection for all timesteps at once (embedding gather fused)
  gemm1_xproj_kernel<<<dim3(16, 80), blk, 0, stream>>>(
      features, captions, embed_W, W_ih, b_ih, b_hh, Gx);

  // Sequential recurrence: 20 (GEMM + pointwise) pairs
  for (int t = 0; t < T_; ++t) {
    float* Gt = Gx + (size_t)t * B_ * G4H;
    gemm2_hproj_kernel<<<dim3(16, 4), blk, 0, stream>>>(h_cur, W_hh, Gt);
    lstm_pointwise_kernel<<<dim3((B_ * H_) / 256), blk, 0, stream>>>(
        Gt, h_cur, c_cur, hs + (size_t)t * B_ * H_);
  }

  // Vocab projection with packed predicated store
  gemm3_vocab_kernel<<<dim3(40, 80), blk, 0, stream>>>(hs, lin_W, lin_b, cnts,
                                                       offs, out);
}